// SAW_53395033424216
// MI455X (gfx1250) — compile-verified
//
#include <hip/hip_runtime.h>
#include <hip/hip_bf16.h>
#include <math.h>

typedef __attribute__((ext_vector_type(2))) float v2f;
typedef __attribute__((ext_vector_type(8))) float v8f;

#define CH     512      // input channels
#define BATCH  16       // B
#define CGRP   16       // C (classes per group / tile size)
#define NGRP   32       // G = CH / CGRP
#define HWTOT  16384    // H*W
#define SPLITK 4        // K-split blocks per (b,g)
#define NBG    (BATCH * NGRP)   // 512 Gram tiles

// ---------------------------------------------------------------------------
// Phase 0: per selected class, take top-32 channels by |w| (descending, stable
// tie-break like jnp.argsort(-|w|)), emit permutation + sigmoid weights.
// chan[g*16 + j] = idx[sel[j]][g],  wgh[g*16 + j] = sigmoid(|w[sel[j], chan]|)
// ---------------------------------------------------------------------------
__global__ __launch_bounds__(256) void SAW_prep_kernel(
    const float* __restrict__ w, const int* __restrict__ sel,
    int* __restrict__ chan, float* __restrict__ wgh) {
  __shared__ float vals[CGRP][CH];
  for (int i = threadIdx.x; i < CGRP * CH; i += blockDim.x) {
    int j = i >> 9, c = i & (CH - 1);
    vals[j][c] = fabsf(w[(size_t)sel[j] * CH + c]);
  }
  __syncthreads();
  int j = threadIdx.x;
  if (j < CGRP) {
    for (int k = 0; k < NGRP; ++k) {
      float best = -1.0f;
      int bi = 0;
      for (int c = 0; c < CH; ++c) {
        float v = vals[j][c];
        if (v > best) { best = v; bi = c; }   // strict > == stable argsort
      }
      vals[j][bi] = -2.0f;                    // knock out, |w| >= 0
      chan[k * CGRP + j] = bi;
      wgh[k * CGRP + j] = 1.0f / (1.0f + expf(-best));
    }
  }
}

// ---------------------------------------------------------------------------
// Phase 1: Gram tiles via V_WMMA_F32_16X16X4_F32.  One block = one (b,g,kslice).
// 8 waves each own a 512-deep K range; A-fragment == B-fragment for X X^T:
//   lane L (0..15):  v2f = X[row=L][k..k+1];  lane 16+L: X[L][k+2..k+3]
// Two accumulator chains hide WMMA latency; LDS cross-wave reduce; partial
// results stored per kslice (no atomics -> deterministic).
// ---------------------------------------------------------------------------
__global__ __launch_bounds__(256) void SAW_gram_kernel(
    const float* __restrict__ x, const int* __restrict__ chan,
    float* __restrict__ covp) {
  const int bg   = blockIdx.x / SPLITK;
  const int ks   = blockIdx.x % SPLITK;
  const int b    = bg >> 5;            // NGRP = 32
  const int g    = bg & (NGRP - 1);
  const int lane = threadIdx.x & 31;
  const int wave = threadIdx.x >> 5;   // 8 waves
  const int row  = lane & 15;
  const int kofs = (lane >> 4) << 1;   // 0 or 2

  const int   ch   = chan[g * CGRP + row];
  const float* base = x + ((size_t)b * CH + ch) * HWTOT;

  const int KB   = HWTOT / SPLITK;     // 4096 per block
  const int KW   = KB / 8;             // 512 per wave
  const int kbeg = ks * KB + wave * KW;

  v8f acc0 = {0.f, 0.f, 0.f, 0.f, 0.f, 0.f, 0.f, 0.f};
  v8f acc1 = {0.f, 0.f, 0.f, 0.f, 0.f, 0.f, 0.f, 0.f};

#pragma unroll 4
  for (int k = 0; k < KW; k += 8) {
    v2f a0 = *(const v2f*)(base + kbeg + k + kofs);
    v2f a1 = *(const v2f*)(base + kbeg + k + 4 + kofs);
    // D = A * A^T + C   (A-frag and B-frag of a Gram are identical registers)
    acc0 = __builtin_amdgcn_wmma_f32_16x16x4_f32(
        false, a0, false, a0, (short)0, acc0, false, false);
    acc1 = __builtin_amdgcn_wmma_f32_16x16x4_f32(
        false, a1, false, a1, (short)0, acc1, false, false);
  }

  // C/D layout: VGPR r, lanes 0-15 -> M=r, N=lane; lanes 16-31 -> M=r+8, N=lane-16
  __shared__ float sc[8][256];
  const int nn = lane & 15;
  const int hi = (lane >> 4) * 8;
#pragma unroll
  for (int r = 0; r < 8; ++r)
    sc[wave][(hi + r) * 16 + nn] = acc0[r] + acc1[r];
  __syncthreads();

  float s = 0.f;
#pragma unroll
  for (int wv = 0; wv < 8; ++wv) s += sc[wv][threadIdx.x];
  covp[((size_t)ks * NBG + bg) * 256 + threadIdx.x] = s;
}

// ---------------------------------------------------------------------------
// Phase 2: loss = sum_{b,g} sum_{i<j} wgh_i*wgh_j*|cov_ij| / ((HW-1)*120*B)
// (eps*I is on the masked-out diagonal; clamp(min=0) is a no-op on sum of |.|;
//  sigmoid weights are positive so they commute with |.|)
// ---------------------------------------------------------------------------
__global__ __launch_bounds__(512) void SAW_reduce_kernel(
    const float* __restrict__ covp, const float* __restrict__ wgh,
    float* __restrict__ out) {
  const int bg = threadIdx.x;          // 0..511
  const int g  = bg & (NGRP - 1);

  float wl[CGRP];
#pragma unroll
  for (int i = 0; i < CGRP; ++i) wl[i] = wgh[g * CGRP + i];

  float s = 0.f;
  for (int i = 0; i < CGRP; ++i) {
    for (int j = i + 1; j < CGRP; ++j) {
      float v = 0.f;
#pragma unroll
      for (int ks = 0; ks < SPLITK; ++ks)
        v += covp[((size_t)ks * NBG + bg) * 256 + i * 16 + j];
      s += wl[i] * wl[j] * fabsf(v);
    }
  }
  s *= 1.0f / ((float)(HWTOT - 1) * 120.0f * (float)BATCH);

  __shared__ float red[512];
  red[threadIdx.x] = s;
  __syncthreads();
  for (int off = 256; off > 0; off >>= 1) {
    if (threadIdx.x < off) red[threadIdx.x] += red[threadIdx.x + off];
    __syncthreads();
  }
  if (threadIdx.x == 0) out[0] = red[0];
}

// ---------------------------------------------------------------------------
extern "C" void kernel_launch(void* const* d_in, const int* in_sizes, int n_in,
                              void* d_out, int out_size, void* d_ws, size_t ws_size,
                              hipStream_t stream) {
  const float* x   = (const float*)d_in[0];   // [16,512,128,128] f32
  const float* w   = (const float*)d_in[1];   // [19,512] f32
  const int*   sel = (const int*)d_in[2];     // [16] i32

  char* wsb = (char*)d_ws;
  const size_t covBytes = (size_t)SPLITK * NBG * 256 * sizeof(float); // 2 MB
  float* covp = (float*)wsb;
  int*   chan = (int*)(wsb + covBytes);
  float* wgh  = (float*)(wsb + covBytes + NBG * sizeof(int));

  SAW_prep_kernel<<<1, 256, 0, stream>>>(w, sel, chan, wgh);
  SAW_gram_kernel<<<NBG * SPLITK, 256, 0, stream>>>(x, chan, covp);
  SAW_reduce_kernel<<<1, 512, 0, stream>>>(covp, wgh, (float*)d_out);
}